// QuantumLayer_42657615184198
// MI455X (gfx1250) — compile-verified
//
#include <hip/hip_runtime.h>

// ---------------------------------------------------------------------------
// 9-qubit (DIM=512) quantum circuit expectation, fused for MI455X (gfx1250).
//   out[b,q] = sum_j z[q,j] * |(U s_b)_j|^2
// Phase (-i)^popc(k) is absorbed into U' so the RHS is real:
//   psi = U' * m   ->  2 real GEMMs on V_WMMA_F32_16X16X4_F32.
// ---------------------------------------------------------------------------

#define QN    9
#define QDIM  512
#define WGCOLS 128          // batch columns per workgroup (8 waves x 16)
#define MSTR  132           // LDS table row stride (132 % 64 != 0 -> no bank conflicts)

typedef __attribute__((ext_vector_type(2))) float v2f;
typedef __attribute__((ext_vector_type(4))) float v4f;
typedef __attribute__((ext_vector_type(8))) float v8f;

// Build-stage U layout (K-major, K pairs interleaved per row):
//   UP[(k>>1)*1024 + m*2 + (k&1)] = U[m][k]   (k = basis column, m = row)
__device__ __forceinline__ int up_idx(int k, int m) {
  return ((k >> 1) << 10) + (m << 1) + (k & 1);
}

// ---------------- kernel 1a: U = identity ----------------
__global__ __launch_bounds__(256) void qk_init(float* __restrict__ ur,
                                               float* __restrict__ ui) {
  int t = blockIdx.x * 256 + threadIdx.x;   // 262144 threads
  int k = t >> 9, j = t & 511;
  int idx = up_idx(k, j);
  ur[idx] = (j == k) ? 1.0f : 0.0f;
  ui[idx] = 0.0f;
}

// ---------------- kernel 1b: apply one single-qubit rotation gate ----------
__global__ __launch_bounds__(256) void qk_gate(float* __restrict__ vr,
                                               float* __restrict__ vi,
                                               const float* __restrict__ w3,
                                               int wq) {
  int t = blockIdx.x * 256 + threadIdx.x;   // 131072 threads
  int k  = t >> 8;                          // column (basis state) 0..511
  int pr = t & 255;                         // pair index 0..255
  int mask = 1 << (8 - wq);
  int low  = pr & (mask - 1);
  int high = (pr & ~(mask - 1)) << 1;
  int j0 = high | low;
  int j1 = j0 | mask;

  float phi = w3[0], th = w3[1], om = w3[2];
  float sh, ch;  sincosf(0.5f * th, &sh, &ch);
  float ap = -0.5f * (phi + om);            // m00 = e^{i ap} c,  m11 = e^{-i ap} c
  float am = -0.5f * (phi - om);            // m10 = e^{i am} s,  m01 = -e^{-i am} s
  float sap, cap; sincosf(ap, &sap, &cap);
  float sam, cam; sincosf(am, &sam, &cam);
  float m00r =  cap * ch, m00i =  sap * ch;
  float m01r = -cam * sh, m01i =  sam * sh;
  float m10r =  cam * sh, m10i =  sam * sh;
  float m11r =  cap * ch, m11i = -sap * ch;

  int i0 = up_idx(k, j0), i1 = up_idx(k, j1);
  float v0r = vr[i0], v0i = vi[i0], v1r = vr[i1], v1i = vi[i1];
  float n0r = m00r * v0r - m00i * v0i + m01r * v1r - m01i * v1i;
  float n0i = m00r * v0i + m00i * v0r + m01r * v1i + m01i * v1r;
  float n1r = m10r * v0r - m10i * v0i + m11r * v1r - m11i * v1i;
  float n1i = m10r * v0i + m10i * v0r + m11r * v1i + m11i * v1r;
  vr[i0] = n0r; vi[i0] = n0i;
  vr[i1] = n1r; vi[i1] = n1i;
}

// ---------------- kernel 1c: CNOT-ring layer = row permutation -------------
__global__ __launch_bounds__(256) void qk_perm(const float* __restrict__ vr_in,
                                               const float* __restrict__ vi_in,
                                               float* __restrict__ vr_out,
                                               float* __restrict__ vi_out,
                                               int r) {
  int t = blockIdx.x * 256 + threadIdx.x;   // 262144 threads
  int k = t >> 9, j = t & 511;
  int cur = j;
#pragma unroll
  for (int i = 0; i < 9; ++i) {
    int bit = (cur >> (8 - i)) & 1;
    int tq  = (i + r) % 9;
    cur ^= bit << (8 - tq);
  }
  vr_out[up_idx(k, cur)] = vr_in[up_idx(k, j)];
  vi_out[up_idx(k, cur)] = vi_in[up_idx(k, j)];
}

// ---------------- kernel 1d: absorb (-i)^popc(k) phase + pack for b128 -----
// Packed layout consumed by the GEMM:
//   UP4[((k>>1)*512 + m)*4 + {0,1,2,3}] = { r(k), r(k+1), i(k), i(k+1) }
__global__ __launch_bounds__(256) void qk_pack(const float* __restrict__ vr,
                                               const float* __restrict__ vi,
                                               float* __restrict__ up4) {
  int t = blockIdx.x * 256 + threadIdx.x;   // 262144 threads
  int k = t >> 9, m = t & 511;
  float rr = vr[up_idx(k, m)];
  float ii = vi[up_idx(k, m)];
  int c = __popc(k) & 3;                    // multiply by (-i)^c
  float nr, ni;
  if      (c == 0) { nr =  rr; ni =  ii; }
  else if (c == 1) { nr =  ii; ni = -rr; }
  else if (c == 2) { nr = -rr; ni = -ii; }
  else             { nr = -ii; ni =  rr; }
  int base = (((k >> 1) << 9) + m) << 2;
  int q = k & 1;
  up4[base + q]     = nr;
  up4[base + 2 + q] = ni;
}

// ---------------- kernel 2: fused Psi = U'*m, probs, Z reduction -----------
__global__ __launch_bounds__(256) void qk_main(const float* __restrict__ x,
                                               const float* __restrict__ up4,
                                               float* __restrict__ out,
                                               int batch) {
  __shared__ float ldsMhi[32 * MSTR];       // per-column high-bit products
  __shared__ float ldsMlo[16 * MSTR];       // per-column low-bit products

  const int tid  = threadIdx.x;
  const int lane = tid & 31;
  const int wave = tid >> 5;
  const int col_base = blockIdx.x * WGCOLS;

  // --- prologue: per-column Kronecker factor tables ---
  if (tid < WGCOLS) {
    int b = col_base + tid;
    if (b >= batch) b = batch - 1;          // clamp (reads only)
    float cw[9], sw[9];
#pragma unroll
    for (int w = 0; w < 9; ++w)
      sincosf(0.5f * x[b * 9 + w], &sw[w], &cw[w]);
    for (int khi = 0; khi < 32; ++khi) {
      float p = 1.0f;
#pragma unroll
      for (int w = 0; w < 5; ++w)
        p *= ((khi >> (4 - w)) & 1) ? sw[w] : cw[w];
      ldsMhi[khi * MSTR + tid] = p;
    }
    for (int klo = 0; klo < 16; ++klo) {
      float p = 1.0f;
#pragma unroll
      for (int w = 5; w < 9; ++w)
        p *= ((klo >> (8 - w)) & 1) ? sw[w] : cw[w];
      ldsMlo[klo * MSTR + tid] = p;
    }
  }
  __syncthreads();

  const int arow    = lane & 15;            // A-fragment row within tile
  const int koff    = (lane >> 4) << 1;     // K offset: 0 (lanes 0-15), 2 (16-31)
  const int n_local = (wave << 4) + (lane & 15);  // column index within WG

  // Preload this lane's mlo pairs (fixed for whole kernel):
  //   K-step j inside a 16-K block touches klo = koff+4j, koff+4j+1.
  v2f mlop[4];
#pragma unroll
  for (int j = 0; j < 4; ++j) {
    mlop[j].x = ldsMlo[(koff + 4 * j)     * MSTR + n_local];
    mlop[j].y = ldsMlo[(koff + 4 * j + 1) * MSTR + n_local];
  }

  float accq[9];
#pragma unroll
  for (int q = 0; q < 9; ++q) accq[q] = 0.0f;

  for (int mt = 0; mt < 32; ++mt) {
    const int m_base = mt << 4;
    __syncthreads();                        // keep waves converged -> WGP$ hits on U

    v8f acc_r = {};
    v8f acc_i = {};

    // base offset of this lane's first A fragment: kpair = koff>>1, row m_base+arow
    const float* up = up4 + (((koff >> 1) << 9) + m_base + arow) * 4;

#pragma unroll 2
    for (int k16 = 0; k16 < 32; ++k16) {
      const float mhi = ldsMhi[k16 * MSTR + n_local];
#pragma unroll
      for (int j = 0; j < 4; ++j) {
        // A fragments: one coalesced b128 supplies Ur and Ui pairs (L2-hot)
        v4f u = *(const v4f*)up;
        up += 2 * 2048;                     // k += 4  ->  kpair += 2
        v2f ar = { u.x, u.y };
        v2f ai = { u.z, u.w };
        v2f b = mhi * mlop[j];              // real RHS: one v_pk_mul_f32

        acc_r = __builtin_amdgcn_wmma_f32_16x16x4_f32(false, ar, false, b, (short)0, acc_r, false, false);
        acc_i = __builtin_amdgcn_wmma_f32_16x16x4_f32(false, ai, false, b, (short)0, acc_i, false, false);
      }
    }

    // epilogue for this row tile: probs + Z-diagonal (+/-1) reduction
    const int mrow0 = m_base + ((lane >> 4) << 3);   // D layout: M = r + 8*(lane>>4)
#pragma unroll
    for (int r = 0; r < 8; ++r) {
      float p = acc_r[r] * acc_r[r] + acc_i[r] * acc_i[r];
      int m = mrow0 + r;
#pragma unroll
      for (int q = 0; q < 9; ++q)
        accq[q] = ((m >> (8 - q)) & 1) ? (accq[q] - p) : (accq[q] + p);
    }
  }

  // combine lane halves (same column N = lane&15, disjoint row halves)
#pragma unroll
  for (int q = 0; q < 9; ++q)
    accq[q] += __shfl_xor(accq[q], 16, 32);

  if (lane < 16) {
    int b = col_base + (wave << 4) + lane;
    if (b < batch) {
#pragma unroll
      for (int q = 0; q < 9; ++q)
        out[b * 9 + q] = accq[q];
    }
  }
}

// ---------------------------------------------------------------------------
extern "C" void kernel_launch(void* const* d_in, const int* in_sizes, int n_in,
                              void* d_out, int out_size, void* d_ws, size_t ws_size,
                              hipStream_t stream) {
  const float* x   = (const float*)d_in[0];   // (32768, 9) f32
  const float* wts = (const float*)d_in[1];   // (3, 9, 3)  f32
  float* out = (float*)d_out;                 // (32768, 9) f32
  const int batch = in_sizes[0] / 9;

  // workspace: ping-pong complex U buffers A,B (4 x 1 MB).
  // The build ends in B (3 swaps), so the packed U' (2 MB) reuses A's region.
  const size_t UFL = (size_t)QDIM * QDIM;
  float* base = (float*)d_ws;
  float* Ar = base;             float* Ai = base + UFL;
  float* Br = base + 2 * UFL;   float* Bi = base + 3 * UFL;

  qk_init<<<1024, 256, 0, stream>>>(Ar, Ai);

  float *cr = Ar, *ci = Ai, *nr = Br, *ni = Bi;
  for (int l = 0; l < 3; ++l) {
    for (int w = 0; w < 9; ++w)
      qk_gate<<<512, 256, 0, stream>>>(cr, ci, wts + l * 27 + w * 3, w);
    qk_perm<<<1024, 256, 0, stream>>>(cr, ci, nr, ni, l + 1);  // RANGES = {1,2,3}
    float* t;
    t = cr; cr = nr; nr = t;
    t = ci; ci = ni; ni = t;
  }
  // cr/ci == Br/Bi here; pack U' (phase-absorbed) into the dead A region.
  float* up4 = base;                           // 2*UFL floats
  qk_pack<<<1024, 256, 0, stream>>>(cr, ci, up4);

  const int wgs = (batch + WGCOLS - 1) / WGCOLS;   // 256 workgroups
  qk_main<<<wgs, 256, 0, stream>>>(x, up4, out, batch);
}